// SliddingWindowAttention_15839839388318
// MI455X (gfx1250) — compile-verified
//
#include <hip/hip_runtime.h>
#include <hip/hip_bf16.h>

#define EMBED 768
#define NSEQ  4096
#define DHEAD 64
#define LDSTR 772  // 768 + 4 pad -> conflict-free WMMA A-fragment reads

typedef __attribute__((ext_vector_type(2))) float v2f;
typedef __attribute__((ext_vector_type(8))) float v8f;

// ---------------------------------------------------------------------------
// Kernel 1: fused QKV projection.  One block = 12 waves; each wave computes a
// 16x16 tile of [Q|K|V] for the block's 16 rows using V_WMMA_F32_16X16X4_F32.
// ---------------------------------------------------------------------------
__global__ __launch_bounds__(384) void qkv_gemm_kernel(
    const float* __restrict__ x,
    const float* __restrict__ Wq, const float* __restrict__ bq,
    const float* __restrict__ Wk, const float* __restrict__ bk,
    const float* __restrict__ Wv, const float* __restrict__ bv,
    float* __restrict__ Q, float* __restrict__ K, float* __restrict__ V) {
  __shared__ float lds_x[16 * LDSTR];

  const int m0  = blockIdx.x * 16;   // row tile base in [0, B*N)
  const int tid = threadIdx.x;       // 0..383

  // Cooperative load of the 16x768 x-tile into LDS (float4 vectorized).
  for (int v = tid; v < 16 * EMBED / 4; v += 384) {
    const int r  = v / (EMBED / 4);
    const int c4 = v % (EMBED / 4);
    const float4 val = ((const float4*)(x + (size_t)(m0 + r) * EMBED))[c4];
    float* dst = &lds_x[r * LDSTR + c4 * 4];
    dst[0] = val.x; dst[1] = val.y; dst[2] = val.z; dst[3] = val.w;
  }
  __syncthreads();

  const int wave = tid >> 5;   // 0..11
  const int lane = tid & 31;
  const int ln   = lane & 15;  // N-column within tile / A-row M
  const int lh   = lane >> 4;  // half-wave: K offset (A/B), +8 rows (C/D)

  const float* W; const float* bias; float* outp; int n0;
  if (wave < 4)      { W = Wq; bias = bq; outp = Q; n0 = wave * 16; }
  else if (wave < 8) { W = Wk; bias = bk; outp = K; n0 = (wave - 4) * 16; }
  else               { W = Wv; bias = bv; outp = V; n0 = (wave - 8) * 16; }

  // Accumulator pre-loaded with bias (same bias for all 8 D rows of a lane).
  const float bval = bias[n0 + ln];
  v8f acc;
#pragma unroll
  for (int r = 0; r < 8; ++r) acc[r] = bval;

  const float* lrow = &lds_x[ln * LDSTR];   // A row M = lane%16
  const float* wcol = W + n0 + ln;          // B col N = lane%16

#pragma unroll 4
  for (int k0 = 0; k0 < EMBED; k0 += 4) {
    const int ka = k0 + 2 * lh;                 // fragment K base for this half
    v2f a;  a[0] = lrow[ka];                    // A[m][ka], A[m][ka+1]
            a[1] = lrow[ka + 1];
    v2f bf; bf[0] = wcol[(size_t)ka * DHEAD];   // B[ka][n], B[ka+1][n]
            bf[1] = wcol[(size_t)(ka + 1) * DHEAD];
    acc = __builtin_amdgcn_wmma_f32_16x16x4_f32(
        /*neg_a=*/false, a, /*neg_b=*/false, bf,
        /*c_mod=*/(short)0, acc, /*reuse_a=*/false, /*reuse_b=*/false);
  }

  // D layout: vgpr r -> row (r + 8*lh), col = lane%16.
#pragma unroll
  for (int r = 0; r < 8; ++r) {
    const int m = m0 + r + 8 * lh;
    outp[(size_t)m * DHEAD + n0 + ln] = acc[r];
  }
}

// ---------------------------------------------------------------------------
// Kernel 2: per-batch Srest[b,d] = sum_{j=3}^{N-1} V[b,j,d].
// Deterministic tree reduction (no float atomics).
// ---------------------------------------------------------------------------
__global__ __launch_bounds__(256) void vsum_kernel(const float* __restrict__ V,
                                                   float* __restrict__ Srest) {
  __shared__ float red[256];
  const int b = blockIdx.x;       // batch
  const int t = threadIdx.x;      // 0..255
  const int d = t & 63;
  const int s = t >> 6;           // 4-way row slicing
  float acc = 0.0f;
  for (int j = 3 + s; j < NSEQ; j += 4)
    acc += V[((size_t)b * NSEQ + j) * DHEAD + d];
  red[t] = acc;
  __syncthreads();
  if (s == 0)
    Srest[b * DHEAD + d] = red[d] + red[64 + d] + red[128 + d] + red[192 + d];
}

// ---------------------------------------------------------------------------
// Kernel 3: per-row closed-form softmax combine.  One wave32 per row; each
// lane owns 2 of the 64 head dims; dot products reduced with __shfl_xor.
// ---------------------------------------------------------------------------
__global__ __launch_bounds__(256) void combine_kernel(
    const float* __restrict__ Q, const float* __restrict__ K,
    const float* __restrict__ V, const float* __restrict__ Srest,
    float* __restrict__ out) {
  const int tid  = threadIdx.x;
  const int wave = tid >> 5;
  const int lane = tid & 31;
  const int g = blockIdx.x * 8 + wave;  // global row in [0, B*N)
  const int b = g >> 12;                // / NSEQ
  const int i = g & (NSEQ - 1);         // % NSEQ

  const float2 q = ((const float2*)(Q + (size_t)g * DHEAD))[lane];

  float pa = 0.0f, pb = 0.0f, pc = 0.0f;
  {
    const float2 kc = ((const float2*)(K + (size_t)g * DHEAD))[lane];
    pb = q.x * kc.x + q.y * kc.y;
  }
  if (i > 0) {
    const float2 kp = ((const float2*)(K + (size_t)(g - 1) * DHEAD))[lane];
    pa = q.x * kp.x + q.y * kp.y;
  }
  if (i < NSEQ - 1) {
    const float2 kn = ((const float2*)(K + (size_t)(g + 1) * DHEAD))[lane];
    pc = q.x * kn.x + q.y * kn.y;
  }
#pragma unroll
  for (int off = 16; off > 0; off >>= 1) {
    pa += __shfl_xor(pa, off, 32);
    pb += __shfl_xor(pb, off, 32);
    pc += __shfl_xor(pc, off, 32);
  }
  const float a  = (i > 0) ? pa : 0.0f;          // columns[i,0] stays 0 at edge
  const float bs = pb;
  const float c  = (i < NSEQ - 1) ? pc : 0.0f;   // columns[i,2] stays 0 at edge

  const float m  = fmaxf(fmaxf(a, bs), fmaxf(c, 0.0f));
  const float ea = expf(a - m);
  const float eb = expf(bs - m);
  const float ec = expf(c - m);
  const float ez = expf(-m);                     // each of the N-3 zero columns
  const float inv = 1.0f / (ea + eb + ec + (float)(NSEQ - 3) * ez);

  const float* Vb = V + (size_t)b * NSEQ * DHEAD;
  const float2 v0 = ((const float2*)(Vb))[lane];
  const float2 v1 = ((const float2*)(Vb + DHEAD))[lane];
  const float2 v2 = ((const float2*)(Vb + 2 * DHEAD))[lane];
  const float2 sr = ((const float2*)(Srest + b * DHEAD))[lane];

  float2 o;
  o.x = (ea * v0.x + eb * v1.x + ec * v2.x + ez * sr.x) * inv;
  o.y = (ea * v0.y + eb * v1.y + ec * v2.y + ez * sr.y) * inv;
  ((float2*)(out + (size_t)g * DHEAD))[lane] = o;
}

// ---------------------------------------------------------------------------
extern "C" void kernel_launch(void* const* d_in, const int* in_sizes, int n_in,
                              void* d_out, int out_size, void* d_ws, size_t ws_size,
                              hipStream_t stream) {
  const float* x  = (const float*)d_in[0];
  const float* Wq = (const float*)d_in[1];
  const float* bq = (const float*)d_in[2];
  const float* Wk = (const float*)d_in[3];
  const float* bk = (const float*)d_in[4];
  const float* Wv = (const float*)d_in[5];
  const float* bv = (const float*)d_in[6];
  float* out = (float*)d_out;

  const int BN = in_sizes[0] / EMBED;  // B*N = 16384
  const int B  = BN / NSEQ;            // 4

  float* Q     = (float*)d_ws;
  float* K     = Q + (size_t)BN * DHEAD;
  float* V     = K + (size_t)BN * DHEAD;
  float* Srest = V + (size_t)BN * DHEAD;

  qkv_gemm_kernel<<<BN / 16, 384, 0, stream>>>(x, Wq, bq, Wk, bk, Wv, bv, Q, K, V);
  vsum_kernel<<<B, 256, 0, stream>>>(V, Srest);
  combine_kernel<<<BN / 8, 256, 0, stream>>>(Q, K, V, Srest, out);
}